// Net_64433099375363
// MI455X (gfx1250) — compile-verified
//
#include <hip/hip_runtime.h>
#include <math.h>

#define INF  512
#define HID  8
#define NCLS 16
#define TB   256

// GEMM1 async-staging config
#define G1_WAVES 4
#define G1_TB    (G1_WAVES * 32)
#define KC       32                 // K-chunk (floats) staged per buffer
#define ROWP     (KC + 4)           // padded LDS row stride -> conflict-free A reads
#define NCHUNK   (INF / KC)         // 16 chunks

typedef __attribute__((ext_vector_type(2))) float v2f;
typedef __attribute__((ext_vector_type(8))) float v8f;

// -------- CDNA5 async global->LDS copy (ASYNCcnt-tracked) --------

__device__ __forceinline__ void async_copy_b128(unsigned lds_byte, const void* gaddr) {
  asm volatile("global_load_async_to_lds_b128 %0, %1, off"
               :: "v"(lds_byte), "v"(gaddr)
               : "memory");
}
__device__ __forceinline__ void wait_async_le4() {
  asm volatile("s_wait_asynccnt 0x4" ::: "memory");
}
__device__ __forceinline__ void wait_async_le0() {
  asm volatile("s_wait_asynccnt 0x0" ::: "memory");
}

// ---------------- degree / normalization ----------------

__global__ void k_init_deg(unsigned* __restrict__ deg, int N) {
  int i = blockIdx.x * TB + threadIdx.x;
  if (i < N) deg[i] = 1u;                       // self-loop contributes 1 to row-degree
}

__global__ void k_count_deg(const int* __restrict__ ei, unsigned* __restrict__ deg, int E) {
  int e = blockIdx.x * TB + threadIdx.x;
  if (e < E) atomicAdd(&deg[ei[e]], 1u);        // degree over row (source) indices
}

__global__ void k_dinv(const unsigned* __restrict__ deg, float* __restrict__ dinv, int N) {
  int i = blockIdx.x * TB + threadIdx.x;
  if (i < N) dinv[i] = rsqrtf((float)deg[i]);
}

// ---- GEMM1: h1[N,8] = x[N,512] @ W1^T + b1 (WMMA f32 16x16x4, async-LDS staged) ----

__global__ void __launch_bounds__(G1_TB)
k_gemm1(const float* __restrict__ x, const float* __restrict__ W1,
        const float* __restrict__ b1, float* __restrict__ h1, int N) {
  __shared__ float sW[INF * 16];                       // B = W1^T padded to 16 cols (32 KB)
  __shared__ float sA[G1_WAVES][2][16 * ROWP];         // per-wave double buffer (18 KB)

  for (int i = threadIdx.x; i < INF * 16; i += G1_TB) {
    int k = i >> 4, n = i & 15;
    sW[i] = (n < HID) ? W1[n * INF + k] : 0.0f;
  }
  __syncthreads();                                     // sW ready; sA is per-wave private

  const int lane = threadIdx.x & 31;                   // wave32
  const int wv   = threadIdx.x >> 5;
  const int tile = blockIdx.x * G1_WAVES + wv;
  const int base = tile * 16;
  if (base >= N) return;

  const int n  = lane & 15;                            // A: M=lane&15 ; B/C/D: N=lane&15
  const int hi = lane >> 4;
  const int ko = hi * 2;                               // half-wave K pair (0-1 vs 2-3)

  // staging map: one b128/lane; 8 lanes cover one row's 128 B chunk (fully coalesced)
  const int ld_row = lane >> 3;                        // 0..3 (+4*j)
  const int ld_col = (lane & 7) * 4;                   // float offset within chunk

  auto issue = [&](int q, int b) {
    const unsigned lbase = (unsigned)(uintptr_t)(&sA[wv][b][0]);
#pragma unroll
    for (int j = 0; j < 4; ++j) {
      int row  = j * 4 + ld_row;
      int grow = base + row; if (grow > N - 1) grow = N - 1;   // clamp tail rows
      const float* gp = x + (size_t)grow * INF + q * KC + ld_col;
      unsigned lds = lbase + (unsigned)((row * ROWP + ld_col) * 4);
      async_copy_b128(lds, gp);
    }
  };

  issue(0, 0);

  const int rA = n;                                    // this lane's A row
  v8f c = {0.f, 0.f, 0.f, 0.f, 0.f, 0.f, 0.f, 0.f};

  for (int q = 0; q < NCHUNK; ++q) {
    if (q + 1 < NCHUNK) { issue(q + 1, (q + 1) & 1); wait_async_le4(); }
    else                { wait_async_le0(); }
    const float* __restrict__ ab = &sA[wv][q & 1][0];
    const float* __restrict__ wb = &sW[(q * KC) * 16];
#pragma unroll
    for (int kk = 0; kk < KC; kk += 4) {
      v2f a, b;
      a.x = ab[rA * ROWP + kk + ko];
      a.y = ab[rA * ROWP + kk + ko + 1];
      b.x = wb[(kk + ko) * 16 + n];
      b.y = wb[(kk + ko + 1) * 16 + n];
      c = __builtin_amdgcn_wmma_f32_16x16x4_f32(false, a, false, b, (short)0, c, false, false);
    }
  }

  if (n < HID) {                                       // only first 8 output cols are real
    const float bias = b1[n];
#pragma unroll
    for (int r = 0; r < 8; ++r) {
      int row = base + hi * 8 + r;                     // C/D: M = vgpr + 8*(lane>>4)
      if (row < N) h1[(size_t)row * HID + n] = c[r] + bias;
    }
  }
}

// ---------------- layer-1 aggregation ----------------

__global__ void k_selfinit8(const float* __restrict__ dinv, const float* __restrict__ h1,
                            float* __restrict__ agg, int N) {
  int i = blockIdx.x * TB + threadIdx.x;
  if (i >= N) return;
  float w = dinv[i] * dinv[i];
  const float4* s = (const float4*)(h1 + (size_t)i * HID);
  float4 a = s[0], b = s[1];
  float4* d = (float4*)(agg + (size_t)i * HID);
  d[0] = make_float4(w * a.x, w * a.y, w * a.z, w * a.w);
  d[1] = make_float4(w * b.x, w * b.y, w * b.z, w * b.w);
}

__global__ void k_scatter8(const int* __restrict__ ei, const float* __restrict__ dinv,
                           const float* __restrict__ h1, float* __restrict__ agg, int E) {
  int e = blockIdx.x * TB + threadIdx.x;
  if (e >= E) return;
  int r = ei[e], c = ei[E + e];
  float w = dinv[r] * dinv[c];
  const float4* s = (const float4*)(h1 + (size_t)r * HID);   // L2-resident gather
  float4 a = s[0], b = s[1];
  float* d = agg + (size_t)c * HID;
  unsafeAtomicAdd(d + 0, w * a.x);
  unsafeAtomicAdd(d + 1, w * a.y);
  unsafeAtomicAdd(d + 2, w * a.z);
  unsafeAtomicAdd(d + 3, w * a.w);
  unsafeAtomicAdd(d + 4, w * b.x);
  unsafeAtomicAdd(d + 5, w * b.y);
  unsafeAtomicAdd(d + 6, w * b.z);
  unsafeAtomicAdd(d + 7, w * b.w);
}

// ---- GEMM2: h2[N,16] = relu(agg1) @ W2^T + b2 (WMMA f32, K=8, fused ReLU) ----

__global__ void __launch_bounds__(TB)
k_gemm2(const float* __restrict__ agg1, const float* __restrict__ W2,
        const float* __restrict__ b2, float* __restrict__ h2, int N) {
  const int lane = threadIdx.x & 31;
  const int tile = blockIdx.x * (TB / 32) + (threadIdx.x >> 5);
  const int base = tile * 16;
  if (base >= N) return;

  const int n  = lane & 15;
  const int hi = lane >> 4;
  const int ko = hi * 2;
  int arow = base + n;
  if (arow > N - 1) arow = N - 1;
  const float* __restrict__ ar = agg1 + (size_t)arow * HID;

  v8f c = {0.f, 0.f, 0.f, 0.f, 0.f, 0.f, 0.f, 0.f};
#pragma unroll
  for (int k0 = 0; k0 < HID; k0 += 4) {
    v2f a, b;
    a.x = fmaxf(ar[k0 + ko], 0.0f);             // fused ReLU
    a.y = fmaxf(ar[k0 + ko + 1], 0.0f);
    b.x = W2[n * HID + k0 + ko];                // B[k][n] = W2[n][k] (tiny, cached)
    b.y = W2[n * HID + k0 + ko + 1];
    c = __builtin_amdgcn_wmma_f32_16x16x4_f32(false, a, false, b, (short)0, c, false, false);
  }

  const float bias = b2[n];
#pragma unroll
  for (int r = 0; r < 8; ++r) {
    int row = base + hi * 8 + r;
    if (row < N) h2[(size_t)row * NCLS + n] = c[r] + bias;
  }
}

// ---------------- layer-2 aggregation (into d_out) ----------------

__global__ void k_selfinit16(const float* __restrict__ dinv, const float* __restrict__ h2,
                             float* __restrict__ agg, int N) {
  int i = blockIdx.x * TB + threadIdx.x;
  if (i >= N) return;
  float w = dinv[i] * dinv[i];
  const float4* s = (const float4*)(h2 + (size_t)i * NCLS);
  float4* d = (float4*)(agg + (size_t)i * NCLS);
#pragma unroll
  for (int q = 0; q < 4; ++q) {
    float4 v = s[q];
    d[q] = make_float4(w * v.x, w * v.y, w * v.z, w * v.w);
  }
}

__global__ void k_scatter16(const int* __restrict__ ei, const float* __restrict__ dinv,
                            const float* __restrict__ h2, float* __restrict__ agg, int E) {
  int e = blockIdx.x * TB + threadIdx.x;
  if (e >= E) return;
  int r = ei[e], c = ei[E + e];
  float w = dinv[r] * dinv[c];
  const float4* s = (const float4*)(h2 + (size_t)r * NCLS);
  float* d = agg + (size_t)c * NCLS;
#pragma unroll
  for (int q = 0; q < 4; ++q) {
    float4 v = s[q];
    unsafeAtomicAdd(d + 4 * q + 0, w * v.x);
    unsafeAtomicAdd(d + 4 * q + 1, w * v.y);
    unsafeAtomicAdd(d + 4 * q + 2, w * v.z);
    unsafeAtomicAdd(d + 4 * q + 3, w * v.w);
  }
}

// ---------------- in-place log_softmax over 16 classes ----------------

__global__ void k_logsoftmax(float* __restrict__ out, int N) {
  int i = blockIdx.x * TB + threadIdx.x;
  if (i >= N) return;
  float* p = out + (size_t)i * NCLS;
  float v[NCLS];
  float m = -INFINITY;
#pragma unroll
  for (int f = 0; f < NCLS; ++f) { v[f] = p[f]; m = fmaxf(m, v[f]); }
  float s = 0.0f;
#pragma unroll
  for (int f = 0; f < NCLS; ++f) s += expf(v[f] - m);
  float l = logf(s) + m;
#pragma unroll
  for (int f = 0; f < NCLS; ++f) p[f] = v[f] - l;
}

// ---------------- launcher ----------------

extern "C" void kernel_launch(void* const* d_in, const int* in_sizes, int n_in,
                              void* d_out, int out_size, void* d_ws, size_t ws_size,
                              hipStream_t stream) {
  const float* x  = (const float*)d_in[0];
  const int*   ei = (const int*)d_in[1];
  const float* W1 = (const float*)d_in[2];
  const float* b1 = (const float*)d_in[3];
  const float* W2 = (const float*)d_in[4];
  const float* b2 = (const float*)d_in[5];
  float* out = (float*)d_out;

  const int N = in_sizes[0] / INF;
  const int E = in_sizes[1] / 2;

  auto align256 = [](size_t v) { return (v + 255) & ~(size_t)255; };
  char* ws = (char*)d_ws;
  size_t o = 0;
  unsigned* deg  = (unsigned*)(ws + o); o = align256(o + (size_t)N * 4);
  float*    dinv = (float*)(ws + o);    o = align256(o + (size_t)N * 4);
  float*    h1   = (float*)(ws + o);    o = align256(o + (size_t)N * HID * 4);
  float*    agg1 = (float*)(ws + o);    o = align256(o + (size_t)N * HID * 4);
  float*    h2   = (float*)(ws + o);    o = align256(o + (size_t)N * NCLS * 4);
  (void)ws_size; (void)n_in; (void)out_size;

  const int nbN = (N + TB - 1) / TB;
  const int nbE = (E + TB - 1) / TB;
  const int tiles = (N + 15) / 16;
  const int nbG1 = (tiles + G1_WAVES - 1) / G1_WAVES;
  const int nbG2 = (tiles + (TB / 32) - 1) / (TB / 32);

  k_init_deg  <<<nbN, TB, 0, stream>>>(deg, N);
  k_count_deg <<<nbE, TB, 0, stream>>>(ei, deg, E);
  k_dinv      <<<nbN, TB, 0, stream>>>(deg, dinv, N);

  k_gemm1     <<<nbG1, G1_TB, 0, stream>>>(x, W1, b1, h1, N);
  k_selfinit8 <<<nbN, TB, 0, stream>>>(dinv, h1, agg1, N);
  k_scatter8  <<<nbE, TB, 0, stream>>>(ei, dinv, h1, agg1, E);

  k_gemm2     <<<nbG2, TB, 0, stream>>>(agg1, W2, b2, h2, N);
  k_selfinit16<<<nbN, TB, 0, stream>>>(dinv, h2, out, N);
  k_scatter16 <<<nbE, TB, 0, stream>>>(ei, dinv, h2, out, E);

  k_logsoftmax<<<nbN, TB, 0, stream>>>(out, N);
}